// RosterGNN_9560597201316
// MI455X (gfx1250) — compile-verified
//
#include <hip/hip_runtime.h>

typedef __attribute__((ext_vector_type(2))) float v2f;
typedef __attribute__((ext_vector_type(8))) float v8f;

#define HDIM 256   // hidden width (both layers output 256)

// ---------------------------------------------------------------------------
// Degree / normalization: deg[i] = 1 (self loop) + #edges with dst==i,
// then dinv[i] = rsqrt(deg[i])  (deg >= 1 always, so no zero guard needed)
// ---------------------------------------------------------------------------
__global__ void deg_init_kernel(float* __restrict__ deg, int n) {
  int i = blockIdx.x * blockDim.x + threadIdx.x;
  if (i < n) deg[i] = 1.0f;
}

__global__ void deg_count_kernel(const int* __restrict__ dst, float* __restrict__ deg, int e) {
  int i = blockIdx.x * blockDim.x + threadIdx.x;
  if (i < e) atomicAdd(&deg[dst[i]], 1.0f);
}

__global__ void deg_rsqrt_kernel(float* __restrict__ deg, int n) {
  int i = blockIdx.x * blockDim.x + threadIdx.x;
  if (i < n) deg[i] = rsqrtf(deg[i]);
}

// ---------------------------------------------------------------------------
// GEMM: out[N,256] = (relu_in ? relu(A) : A)[N,K] @ B[K,256], fp32 WMMA.
// One wave per 16(M) x 64(N) strip: 4 accumulators, A-frag reused 4x per k-step.
// Wave-uniform guard keeps EXEC all-ones inside v_wmma (ISA requirement).
// ---------------------------------------------------------------------------
template <bool RELU_IN>
__global__ void gemm_wmma_f32_kernel(const float* __restrict__ A,
                                     const float* __restrict__ B,
                                     float* __restrict__ out,
                                     int nrows, int K) {
  const int wave = (blockIdx.x * blockDim.x + threadIdx.x) >> 5;
  const int lane = threadIdx.x & 31;
  const int mtiles = nrows >> 4;          // 16-row tiles
  const int njobs  = mtiles * (HDIM / 64);
  if (wave >= njobs) return;              // uniform across the wave

  const int mt = wave >> 2;               // which 16-row tile
  const int ng = wave & 3;                // which 64-col group
  const int rowbase = mt << 4;
  const int colbase = ng << 6;

  const int mrow = lane & 15;             // A: M index / B: N index
  const int koff = (lane >> 4) << 1;      // A/B: K sub-offset {0,2}

  v8f acc0 = {}, acc1 = {}, acc2 = {}, acc3 = {};

  const float* arow = A + (size_t)(rowbase + mrow) * K;

  for (int k0 = 0; k0 < K; k0 += 4) {
    v2f a;
    a.x = arow[k0 + koff];
    a.y = arow[k0 + koff + 1];
    if (RELU_IN) {
      a.x = fmaxf(a.x, 0.0f);
      a.y = fmaxf(a.y, 0.0f);
    }
    const float* b0 = B + (size_t)(k0 + koff) * HDIM + colbase + mrow;
    const float* b1 = b0 + HDIM;          // K+1 row
#pragma unroll
    for (int j = 0; j < 4; ++j) {
      v2f b;
      b.x = b0[j * 16];
      b.y = b1[j * 16];
      v8f* accp = (j == 0) ? &acc0 : (j == 1) ? &acc1 : (j == 2) ? &acc2 : &acc3;
      *accp = __builtin_amdgcn_wmma_f32_16x16x4_f32(
          /*neg_a=*/false, a, /*neg_b=*/false, b,
          /*c_mod=*/(short)0, *accp, /*reuse_a=*/false, /*reuse_b=*/false);
    }
  }

  // C/D layout: VGPR r -> M = r (lanes 0-15) or M = r+8 (lanes 16-31), N = lane&15
  const int crow = (lane >> 4) << 3;
  const int ccol = lane & 15;
  float* obase = out + (size_t)(rowbase + crow) * HDIM + colbase + ccol;
#pragma unroll
  for (int r = 0; r < 8; ++r) {
    obase[(size_t)r * HDIM + 0]  = acc0[r];
    obase[(size_t)r * HDIM + 16] = acc1[r];
    obase[(size_t)r * HDIM + 32] = acc2[r];
    obase[(size_t)r * HDIM + 48] = acc3[r];
  }
}

// ---------------------------------------------------------------------------
// Aggregation init: out[i,f] = dinv[i]^2 * h[i,f] + bias[f]   (self-loop + bias)
// ---------------------------------------------------------------------------
__global__ void agg_init_kernel(const float* __restrict__ h,
                                const float* __restrict__ dinv,
                                const float* __restrict__ bias,
                                float* __restrict__ out, int n) {
  int idx = blockIdx.x * blockDim.x + threadIdx.x;
  if (idx >= n * HDIM) return;
  int i = idx >> 8;          // HDIM == 256
  int f = idx & 255;
  float di = dinv[i];
  out[idx] = di * di * h[idx] + bias[f];
}

// ---------------------------------------------------------------------------
// Edge scatter: one wave per edge. w = dinv[src]*dinv[dst];
// out[dst,:] += w * h[src,:]  — 2x float4 gather + 8 fp32 L2 atomics per lane.
// ---------------------------------------------------------------------------
__global__ void agg_edges_kernel(const int* __restrict__ src,
                                 const int* __restrict__ dst,
                                 const float* __restrict__ dinv,
                                 const float* __restrict__ h,
                                 float* __restrict__ out, int e) {
  const int wave = (blockIdx.x * blockDim.x + threadIdx.x) >> 5;
  const int lane = threadIdx.x & 31;
  if (wave >= e) return;
  const int s = src[wave];
  const int d = dst[wave];
  const float w = dinv[s] * dinv[d];
  const float4* hs = (const float4*)(h + (size_t)s * HDIM);
  float* od = out + (size_t)d * HDIM;
#pragma unroll
  for (int j = 0; j < 2; ++j) {
    const int q = lane + j * 32;          // float4 index within the row
    float4 v = hs[q];
    atomicAdd(od + q * 4 + 0, w * v.x);
    atomicAdd(od + q * 4 + 1, w * v.y);
    atomicAdd(od + q * 4 + 2, w * v.z);
    atomicAdd(od + q * 4 + 3, w * v.w);
  }
}

__global__ void relu_kernel(float* __restrict__ p, int count) {
  int i = blockIdx.x * blockDim.x + threadIdx.x;
  if (i < count) p[i] = fmaxf(p[i], 0.0f);
}

// ---------------------------------------------------------------------------
// Orchestration
// ---------------------------------------------------------------------------
extern "C" void kernel_launch(void* const* d_in, const int* in_sizes, int n_in,
                              void* d_out, int out_size, void* d_ws, size_t ws_size,
                              hipStream_t stream) {
  const float* x   = (const float*)d_in[0];   // [N,128]
  const int*   ei  = (const int*)d_in[1];     // [2,E]
  const float* W1  = (const float*)d_in[2];   // [128,256]
  const float* b1  = (const float*)d_in[3];   // [256]
  const float* W2  = (const float*)d_in[4];   // [256,256]
  const float* b2  = (const float*)d_in[5];   // [256]
  float* out = (float*)d_out;                 // [N,256]

  const int FIN = 128;
  const int N = in_sizes[0] / FIN;
  const int E = in_sizes[1] / 2;
  const int* src = ei;
  const int* dst = ei + E;

  // Scratch layout (L2-resident working set): dinv | hbuf | abuf
  float* dinv = (float*)d_ws;
  float* hbuf = dinv + (((size_t)N + 255) & ~(size_t)255);
  float* abuf = hbuf + (size_t)N * HDIM;

  const int TPB = 256;
  const int nh_blocks   = (N * HDIM + TPB - 1) / TPB;
  const int edge_blocks = (E * 32 + TPB - 1) / TPB;   // one wave per edge

  // --- symmetric normalization (shared by both layers) ---
  deg_init_kernel<<<(N + TPB - 1) / TPB, TPB, 0, stream>>>(dinv, N);
  deg_count_kernel<<<(E + TPB - 1) / TPB, TPB, 0, stream>>>(dst, dinv, E);
  deg_rsqrt_kernel<<<(N + TPB - 1) / TPB, TPB, 0, stream>>>(dinv, N);

  // --- layer 1: h1 = x @ W1 ; a1 = scatter(norm * h1[src]) + b1 ---
  {
    int jobs = (N / 16) * (HDIM / 64);
    int blocks = (jobs * 32 + TPB - 1) / TPB;
    gemm_wmma_f32_kernel<false><<<blocks, TPB, 0, stream>>>(x, W1, hbuf, N, FIN);
  }
  agg_init_kernel<<<nh_blocks, TPB, 0, stream>>>(hbuf, dinv, b1, abuf, N);
  agg_edges_kernel<<<edge_blocks, TPB, 0, stream>>>(src, dst, dinv, hbuf, abuf, E);
  // ReLU of layer 1 is fused into the layer-2 GEMM A-load (saves a 102 MB pass)

  // --- layer 2: h2 = relu(a1) @ W2 ; out = relu(scatter(norm * h2[src]) + b2) ---
  {
    int jobs = (N / 16) * (HDIM / 64);
    int blocks = (jobs * 32 + TPB - 1) / TPB;
    gemm_wmma_f32_kernel<true><<<blocks, TPB, 0, stream>>>(abuf, W2, hbuf, N, HDIM);
  }
  agg_init_kernel<<<nh_blocks, TPB, 0, stream>>>(hbuf, dinv, b2, out, N);
  agg_edges_kernel<<<edge_blocks, TPB, 0, stream>>>(src, dst, dinv, hbuf, out, E);
  relu_kernel<<<nh_blocks, TPB, 0, stream>>>(out, N * HDIM);
}